// GatingNetwork_3410204033645
// MI455X (gfx1250) — compile-verified
//
#include <hip/hip_runtime.h>
#include <hip/hip_bf16.h>

typedef __attribute__((ext_vector_type(2))) float v2f;
typedef __attribute__((ext_vector_type(4))) float v4f;
typedef __attribute__((ext_vector_type(8))) float v8f;

#define C_DIM   2048
#define E_DIM   64
#define KCHUNK  128
#define LSTRIDE 68   // LDS row stride for logits tile (dwords): halves hit disjoint banks
#define EPS     1e-12f

// ---------------------------------------------------------------------------
// Prep: per-expert reciprocal column norm + sigmoid(gates). 64 threads.
// ---------------------------------------------------------------------------
__global__ void gating_prep(const float* __restrict__ sim,
                            const float* __restrict__ gates,
                            float* __restrict__ wnr,
                            float* __restrict__ sig) {
    int e = threadIdx.x;   // 0..63, coalesced across k
    float s = 0.0f;
    for (int k = 0; k < C_DIM; ++k) {
        float v = sim[(size_t)k * E_DIM + e];
        s = fmaf(v, v, s);
    }
    wnr[e] = 1.0f / fmaxf(sqrtf(s), EPS);
    sig[e] = 1.0f / (1.0f + expf(-gates[e]));
}

// ---------------------------------------------------------------------------
// Main: fused normalized GEMM (f32 WMMA) + gating epilogue.
// 128 threads = 4 waves; wave w handles rows [blk*64 + w*16, +16) x all 64 experts.
// B is staged in LDS in fragment-native layout: ldsB[kgrp][e][4 k-values], so
// each WMMA B fragment is a single conflict-free ds_load_b64 into an aligned pair.
// ---------------------------------------------------------------------------
__global__ __launch_bounds__(128) void gating_main(
        const float* __restrict__ x,      // [N, 2048]
        const float* __restrict__ sim,    // [2048, 64] raw weights
        const float* __restrict__ wnr,    // [64] 1/max(||w_col||,eps)
        const float* __restrict__ sigg,   // [64] sigmoid(gates)
        float* __restrict__ out,          // probs|logits|mask|gated, each [N,64]
        int Ntok) {

    __shared__ float ldsB[(KCHUNK / 4) * E_DIM * 4];   // 32 KB staged B chunk
    __shared__ float ldsL[4][16 * LSTRIDE];            // per-wave logits tile
    __shared__ float rowGmax[4][16];
    __shared__ float rowSum[4][16];
    __shared__ unsigned rowMLo[4][16];
    __shared__ unsigned rowMHi[4][16];

    const int tid   = threadIdx.x;
    const int lane  = tid & 31;
    const int wave  = tid >> 5;
    const int nl    = lane & 15;       // column-within-tile / row-within-tile id
    const int half  = lane >> 4;       // 0: K+0/K+1, 1: K+2/K+3 fragment half
    const int half2 = half << 1;

    const int rowBase   = blockIdx.x * 64 + wave * 16;
    const float* xrow   = x + (size_t)(rowBase + nl) * C_DIM;

    v8f acc0 = {}, acc1 = {}, acc2 = {}, acc3 = {};
    float ss = 0.0f;                   // partial row sum-of-squares

    for (int chunk = 0; chunk < C_DIM; chunk += KCHUNK) {
        __syncthreads();
        // Stage sim[chunk .. chunk+127][0..63] as 32 k-groups of 4:
        // ldsB[kgrp*256 + e*4 + j] = sim[chunk + kgrp*4 + j][e]
        #pragma unroll
        for (int i = 0; i < 16; ++i) {
            int g    = i * 128 + tid;         // group id, 0..2047
            int kgrp = g >> 6;                // 0..31
            int e    = g & 63;
            const float* sp = sim + (size_t)(chunk + (kgrp << 2)) * E_DIM + e;
            v4f v = { sp[0], sp[E_DIM], sp[2 * E_DIM], sp[3 * E_DIM] };
            *(v4f*)&ldsB[(kgrp << 8) + (e << 2)] = v;
        }
        __syncthreads();

        if (chunk + KCHUNK < C_DIM)
            __builtin_prefetch(xrow + chunk + KCHUNK, 0, 3);

        #pragma unroll 8
        for (int kk = 0; kk < KCHUNK; kk += 4) {
            // A fragment: lane half picks (K, K+1) or (K+2, K+3) of its row.
            v2f a = *(const v2f*)(xrow + chunk + kk + half2);
            ss = fmaf(a.x, a.x, ss);
            ss = fmaf(a.y, a.y, ss);
            // B fragments: one aligned b64 per 16-expert tile, conflict-free.
            const float* bp = &ldsB[((kk >> 2) << 8) + half2];
            v2f b0 = *(const v2f*)(bp + ((0 * 16 + nl) << 2));
            v2f b1 = *(const v2f*)(bp + ((1 * 16 + nl) << 2));
            v2f b2 = *(const v2f*)(bp + ((2 * 16 + nl) << 2));
            v2f b3 = *(const v2f*)(bp + ((3 * 16 + nl) << 2));
            acc0 = __builtin_amdgcn_wmma_f32_16x16x4_f32(false, a, false, b0, (short)0, acc0, false, false);
            acc1 = __builtin_amdgcn_wmma_f32_16x16x4_f32(false, a, false, b1, (short)0, acc1, false, false);
            acc2 = __builtin_amdgcn_wmma_f32_16x16x4_f32(false, a, false, b2, (short)0, acc2, false, false);
            acc3 = __builtin_amdgcn_wmma_f32_16x16x4_f32(false, a, false, b3, (short)0, acc3, false, false);
        }
    }

    // Full row sum-of-squares: combine the two K-phase halves (wave32 xor-16).
    float ssFull = ss + __shfl_xor(ss, 16, 32);
    float rnorm  = 1.0f / fmaxf(sqrtf(ssFull), EPS);   // valid per-lane for row nl

    // Per-column scale/bias for this lane's expert columns.
    float wv0 = wnr[ 0 + nl], wv1 = wnr[16 + nl], wv2 = wnr[32 + nl], wv3 = wnr[48 + nl];
    float sv0 = sigg[ 0 + nl], sv1 = sigg[16 + nl], sv2 = sigg[32 + nl], sv3 = sigg[48 + nl];

    // Scale accumulators into logits and park in LDS (D layout: row m = r + 8*half).
    #pragma unroll
    for (int r = 0; r < 8; ++r) {
        int m = r + (half << 3);
        float rn = __shfl(rnorm, m, 32);
        float* Lr = &ldsL[wave][m * LSTRIDE];
        Lr[ 0 + nl] = acc0[r] * rn * wv0 - sv0;
        Lr[16 + nl] = acc1[r] * rn * wv1 - sv1;
        Lr[32 + nl] = acc2[r] * rn * wv2 - sv2;
        Lr[48 + nl] = acc3[r] * rn * wv3 - sv3;
    }
    __syncthreads();

    // Phase 1: per-row stats (mask bits, top-2 fallback, softmax max & sum).
    if (lane < 16) {
        const float* Lp = &ldsL[wave][lane * LSTRIDE];
        float m1 = -__builtin_inff(), m2 = -__builtin_inff();
        int i1 = 0, i2 = 0;
        unsigned mlo = 0, mhi = 0;
        bool any = false;
        float gmax = -__builtin_inff();
        for (int e = 0; e < 64; ++e) {
            float v = Lp[e];
            if (v > 0.0f) {
                any = true;
                gmax = fmaxf(gmax, v);
                if (e < 32) mlo |= 1u << e; else mhi |= 1u << (e - 32);
            }
            if (v > m1)      { m2 = m1; i2 = i1; m1 = v; i1 = e; }
            else if (v > m2) { m2 = v; i2 = e; }
        }
        if (!any) {   // fallback: activate top-2 logits
            mlo = 0; mhi = 0;
            if (i1 < 32) mlo |= 1u << i1; else mhi |= 1u << (i1 - 32);
            if (i2 < 32) mlo |= 1u << i2; else mhi |= 1u << (i2 - 32);
            gmax = fmaxf(m1, 0.0f);
        }
        float s = 0.0f;
        for (int e = 0; e < 64; ++e) {
            unsigned bit = (e < 32) ? ((mlo >> e) & 1u) : ((mhi >> (e - 32)) & 1u);
            if (bit) s += __expf(fmaxf(Lp[e], 0.0f) - gmax);
        }
        rowGmax[wave][lane] = gmax;
        rowSum [wave][lane] = s;
        rowMLo [wave][lane] = mlo;
        rowMHi [wave][lane] = mhi;
    }
    __syncthreads();

    // Phase 2: coalesced writeback of probs | logits | mask | gated_logits.
    const size_t NE = (size_t)Ntok * E_DIM;
    float* probsO = out;
    float* logitO = out + NE;
    float* maskO  = out + 2 * NE;
    float* gatedO = out + 3 * NE;
    const size_t gbase = (size_t)rowBase * E_DIM;

    #pragma unroll 4
    for (int j = 0; j < 32; ++j) {
        int lin = j * 32 + lane;           // 0 .. 1023 over the 16x64 tile
        int row = lin >> 6;
        int e   = lin & 63;
        float v = ldsL[wave][row * LSTRIDE + e];
        float g = fmaxf(v, 0.0f);
        unsigned bit = (e < 32) ? ((rowMLo[wave][row] >> e) & 1u)
                                : ((rowMHi[wave][row] >> (e - 32)) & 1u);
        float p = bit ? __expf(g - rowGmax[wave][row]) / rowSum[wave][row] : 0.0f;
        probsO[gbase + lin] = p;
        logitO[gbase + lin] = v;
        maskO [gbase + lin] = bit ? 1.0f : 0.0f;
        gatedO[gbase + lin] = g;
    }
}

// ---------------------------------------------------------------------------
extern "C" void kernel_launch(void* const* d_in, const int* in_sizes, int n_in,
                              void* d_out, int out_size, void* d_ws, size_t ws_size,
                              hipStream_t stream) {
    const float* x     = (const float*)d_in[0];   // hidden_states [4,4096,2048]
    const float* sim   = (const float*)d_in[1];   // sim_matrix [2048,64]
    const float* gates = (const float*)d_in[2];   // gates [64]
    float* out = (float*)d_out;

    float* wnr = (float*)d_ws;        // [64]
    float* sig = wnr + E_DIM;         // [64]

    const int Ntok = in_sizes[0] / C_DIM;         // 16384

    gating_prep<<<1, E_DIM, 0, stream>>>(sim, gates, wnr, sig);
    gating_main<<<Ntok / 64, 128, 0, stream>>>(x, sim, wnr, sig, out, Ntok);
}